// Model_37039797960982
// MI455X (gfx1250) — compile-verified
//
#include <hip/hip_runtime.h>

// ---------------------------------------------------------------------------
// Exact algebraic collapse of the reference GNN:
//   * All ops before the final relu are affine in (h, e); the readout is a
//     mean over rows, so sums commute through both mp_layers and the scan.
//   * nbr indices lie in [NS, NT), i.e. the gathered h rows are never
//     modified by mp_layer -> gather sums are shared by both layers.
// Pipeline (all memory-bound, ~404 MB HBM @ 23.3 TB/s ~= 17us):
//   K1: column sums of e  (352 MB stream, non-temporal)        -> Se (32x11)
//   K2: sums of gathered h rows per neighbor slot d            -> Sh (32x5)
//   K3: sums of h[:NS], h[NS:], h2                             -> 15 floats
//   K4: single block: summed scans (wave32 shfl), readout,
//       MLP 234->128->32->1 via V_WMMA_F32_16X16X4_F32 (exact f32)
// ---------------------------------------------------------------------------

typedef float v2f __attribute__((ext_vector_type(2)));
typedef float v4f __attribute__((ext_vector_type(4)));
typedef float v8f __attribute__((ext_vector_type(8)));

#define GNN_NS    250000
#define GNN_D     32
#define GNN_NT    500000
#define ECOLS     352                      // 32*11 floats per edge row
#define E_F4      ((GNN_NS * ECOLS) / 4)   // 22,000,000 float4s

#define K1_BLOCKS 352
#define K1_THREADS 256
#define K1_STRIDE (K1_BLOCKS * K1_THREADS) // 90112 float4s = 1024 e-rows: keeps
                                           // each thread on fixed columns
#define K1_ROWS   1024                     // partial rows of width ECOLS in ws

#define K2_BLOCKS 512
#define K3_BLOCKS 256

// ws layout (floats)
#define WSE_OFF 0
#define WSE_CNT (K1_STRIDE * 4)            // 360448
#define WSG_OFF (WSE_OFF + WSE_CNT)
#define WSG_CNT (K2_BLOCKS * 160)          // 81920
#define WSH_OFF (WSG_OFF + WSG_CNT)
#define WSH_CNT (K3_BLOCKS * 15)           // 3840

// ---------------- K1: streaming column-sum of e ----------------------------
__global__ void k_reduce_e(const v4f* __restrict__ e4, float* __restrict__ wse) {
    int gid = blockIdx.x * blockDim.x + threadIdx.x;
    v4f acc = {0.f, 0.f, 0.f, 0.f};
    // stride is 0 mod 88 float4s => each thread owns 4 fixed columns of the
    // (NS x 352) matrix; non-temporal: do not pollute L2 (h lives there).
    for (int i = gid; i < E_F4; i += K1_STRIDE) {
        v4f v = __builtin_nontemporal_load(&e4[i]);
        acc.x += v.x; acc.y += v.y; acc.z += v.z; acc.w += v.w;
    }
    v4f* out = (v4f*)(wse + (size_t)gid * 4);
    *out = acc;
}

// ---------------- K2: sum of gathered h rows per slot d --------------------
__global__ void k_reduce_gather(const int* __restrict__ nbr,
                                const float* __restrict__ h,
                                float* __restrict__ wsg) {
    const int tid = threadIdx.x;
    const int gsz = gridDim.x * blockDim.x;          // multiple of 32
    int gid = blockIdx.x * blockDim.x + tid;
    // slot i has d = i % 32 == tid % 32 == lane: each lane owns one fixed d.
    float a0 = 0.f, a1 = 0.f, a2 = 0.f, a3 = 0.f, a4 = 0.f;
    for (int i = gid; i < GNN_NS * GNN_D; i += gsz) {
        int idx = __builtin_nontemporal_load(&nbr[i]);  // index stream: NT
        const float* hp = h + (size_t)idx * 5;          // 5MB region: cache in L2
        a0 += hp[0]; a1 += hp[1]; a2 += hp[2]; a3 += hp[3]; a4 += hp[4];
    }
    __shared__ float red[256 * 5];
    red[tid * 5 + 0] = a0; red[tid * 5 + 1] = a1; red[tid * 5 + 2] = a2;
    red[tid * 5 + 3] = a3; red[tid * 5 + 4] = a4;
    __syncthreads();
    if (tid < 160) {                     // tid = d*5 + j
        int d = tid / 5, j = tid % 5;
        float s = 0.f;
        for (int w = 0; w < 8; ++w) s += red[(w * 32 + d) * 5 + j];
        wsg[(size_t)blockIdx.x * 160 + tid] = s;
    }
}

// ---------------- K3: sums of h[:NS], h[NS:], h2 ---------------------------
__global__ void k_reduce_h(const float* __restrict__ h,
                           const float* __restrict__ h2,
                           float* __restrict__ wsh) {
    const int tid = threadIdx.x;
    const int gsz = gridDim.x * blockDim.x;
    int gid = blockIdx.x * blockDim.x + tid;
    float lo[5] = {0,0,0,0,0}, hi[5] = {0,0,0,0,0}, s2[5] = {0,0,0,0,0};
    for (int r = gid; r < GNN_NT; r += gsz) {
        const float* hp  = h  + (size_t)r * 5;
        const float* h2p = h2 + (size_t)r * 5;
        if (r < GNN_NS) { for (int j = 0; j < 5; ++j) lo[j] += hp[j]; }
        else            { for (int j = 0; j < 5; ++j) hi[j] += hp[j]; }
        for (int j = 0; j < 5; ++j) s2[j] += h2p[j];
    }
    __shared__ float red[256 * 15];
    for (int j = 0; j < 5; ++j) {
        red[tid * 15 + j]      = lo[j];
        red[tid * 15 + 5 + j]  = hi[j];
        red[tid * 15 + 10 + j] = s2[j];
    }
    __syncthreads();
    if (tid < 15) {
        float s = 0.f;
        for (int i = 0; i < 256; ++i) s += red[i * 15 + tid];
        wsh[(size_t)blockIdx.x * 15 + tid] = s;
    }
}

// ---------------- K4: final reduce + summed scans + MLP (WMMA f32) ---------
__global__ void __launch_bounds__(256)
k_final(const float* __restrict__ molfeats,
        const float* __restrict__ V1w, const float* __restrict__ V1b,
        const float* __restrict__ E1w, const float* __restrict__ E1b,
        const float* __restrict__ U1w, const float* __restrict__ U1b,
        const float* __restrict__ V2w, const float* __restrict__ V2b,
        const float* __restrict__ E2w, const float* __restrict__ E2b,
        const float* __restrict__ U2w, const float* __restrict__ U2b,
        const float* __restrict__ Rw,  const float* __restrict__ Rb,
        const float* __restrict__ W1,  const float* __restrict__ b1,
        const float* __restrict__ W2,  const float* __restrict__ b2,
        const float* __restrict__ W3,  const float* __restrict__ b3,
        const float* __restrict__ wse, const float* __restrict__ wsg,
        const float* __restrict__ wsh, float* __restrict__ out) {
    const int tid = threadIdx.x;
    __shared__ float sSe[ECOLS], sSh[160], sums[15];
    __shared__ float fVU1[25], fEU1[55], fb1s[5];
    __shared__ float fVU2[25], fEU2[55], fb2s[5];
    __shared__ float sUc1[25], sUc2[25];
    __shared__ float smm1[160], smm2[160];
    __shared__ float hfin[5];
    __shared__ float xv[240];          // x vector, padded to 236 (+slack)
    __shared__ float y1[128], y2[32];

    // --- Stage A: reduce workspace partials (fixed order => deterministic)
    for (int c = tid; c < ECOLS; c += 256) {
        float s = 0.f;
        for (int r = 0; r < K1_ROWS; ++r) s += wse[(size_t)r * ECOLS + c];
        sSe[c] = s;
    }
    for (int c = tid; c < 160; c += 256) {
        float s = 0.f;
        for (int b = 0; b < K2_BLOCKS; ++b) s += wsg[(size_t)b * 160 + c];
        sSh[c] = s;
    }
    if (tid < 15) {
        float s = 0.f;
        for (int b = 0; b < K3_BLOCKS; ++b) s += wsh[(size_t)b * 15 + tid];
        sums[tid] = s;
    }
    if (tid < 25) { sUc1[tid] = U1w[tid]; sUc2[tid] = U2w[tid]; } // Uc = Uw[:5]
    __syncthreads();

    // --- Stage B: fused weights. Uw rows: [0:5)=Uc, [5:10)=Um, [10:21)=Ue
    if (tid < 25) {                       // VwUm = Vw @ Um  (5x5)
        int i = tid / 5, j = tid % 5;
        float s1 = 0.f, s2 = 0.f;
        for (int k = 0; k < 5; ++k) {
            s1 += V1w[i * 5 + k] * U1w[(5 + k) * 5 + j];
            s2 += V2w[i * 5 + k] * U2w[(5 + k) * 5 + j];
        }
        fVU1[tid] = s1; fVU2[tid] = s2;
    } else if (tid >= 32 && tid < 87) {   // EwUe = Ew @ Ue  (11x5)
        int t = tid - 32, i = t / 5, j = t % 5;
        float s1 = 0.f, s2 = 0.f;
        for (int k = 0; k < 11; ++k) {
            s1 += E1w[i * 11 + k] * U1w[(10 + k) * 5 + j];
            s2 += E2w[i * 11 + k] * U2w[(10 + k) * 5 + j];
        }
        fEU1[t] = s1; fEU2[t] = s2;
    } else if (tid >= 96 && tid < 101) {  // bias = Vb@Um + Eb@Ue + Ub  (5)
        int j = tid - 96;
        float s1 = U1b[j], s2 = U2b[j];
        for (int k = 0; k < 5; ++k) {
            s1 += V1b[k] * U1w[(5 + k) * 5 + j];
            s2 += V2b[k] * U2w[(5 + k) * 5 + j];
        }
        for (int k = 0; k < 11; ++k) {
            s1 += E1b[k] * U1w[(10 + k) * 5 + j];
            s2 += E2b[k] * U2w[(10 + k) * 5 + j];
        }
        fb1s[j] = s1; fb2s[j] = s2;
    }
    __syncthreads();

    // Sum over edges of mm_d:  Sh[d]@VwUm + Se[d]@EwUe + NS*bias
    if (tid < 160) {
        int d = tid / 5, j = tid % 5;
        float s1 = (float)GNN_NS * fb1s[j];
        float s2 = (float)GNN_NS * fb2s[j];
        for (int i = 0; i < 5; ++i) {
            float sh = sSh[d * 5 + i];
            s1 += sh * fVU1[i * 5 + j];
            s2 += sh * fVU2[i * 5 + j];
        }
        for (int i = 0; i < 11; ++i) {
            float se = sSe[d * 11 + i];
            s1 += se * fEU1[i * 5 + j];
            s2 += se * fEU2[i * 5 + j];
        }
        smm1[tid] = s1; smm2[tid] = s2;
    }
    __syncthreads();

    // --- Stage C: summed scans, wave 0, wave32 shfl broadcast of C[0..4]
    if (tid < 32) {
        int j = tid < 5 ? tid : 4;
        float uc1[5], uc2[5];
        for (int i = 0; i < 5; ++i) { uc1[i] = sUc1[i * 5 + j]; uc2[i] = sUc2[i * 5 + j]; }
        float C = sums[j];                                  // sum of h[:NS]
        for (int d = 0; d < GNN_D; ++d) {
            float c0 = __shfl(C, 0, 32), c1 = __shfl(C, 1, 32),
                  c2 = __shfl(C, 2, 32), c3 = __shfl(C, 3, 32),
                  c4 = __shfl(C, 4, 32);
            C = c0 * uc1[0] + c1 * uc1[1] + c2 * uc1[2] + c3 * uc1[3] +
                c4 * uc1[4] + smm1[d * 5 + j];
        }
        for (int d = 0; d < GNN_D; ++d) {                   // layer 2 continues
            float c0 = __shfl(C, 0, 32), c1 = __shfl(C, 1, 32),
                  c2 = __shfl(C, 2, 32), c3 = __shfl(C, 3, 32),
                  c4 = __shfl(C, 4, 32);
            C = c0 * uc2[0] + c1 * uc2[1] + c2 * uc2[2] + c3 * uc2[3] +
                c4 * uc2[4] + smm2[d * 5 + j];
        }
        if (tid < 5) hfin[tid] = C + sums[5 + tid];         // + sum h[NS:]
    }
    __syncthreads();

    // --- Stage D: readout mean -> relu -> build x (234, padded to 236)
    if (tid < 32) {
        float s = 0.f;
        for (int i = 0; i < 5; ++i)
            s += hfin[i] * Rw[i * 32 + tid] + sums[10 + i] * Rw[(5 + i) * 32 + tid];
        float ro = s / (float)GNN_NT + Rb[tid];
        xv[tid] = ro > 0.f ? ro : 0.f;
    }
    for (int m = tid; m < 202; m += 256) {
        float v = molfeats[m];
        xv[32 + m] = v > 0.f ? v : 0.f;
    }
    if (tid < 6) xv[234 + tid] = 0.f;                        // K padding
    __syncthreads();

    // --- Stage E: MLP via V_WMMA_F32_16X16X4_F32 (exact f32)
    // A (16x4): lanes 0-15 rows M0-15 K={kb,kb+1}=0,1; lanes 16-31 K=2,3.
    //           x replicated across all M rows -> use D row 0 (vgpr0,lanes0-15).
    // B (4x16): lane n = lane&15, same K split.
    const int lane = tid & 31, wave = tid >> 5;
    const int kb = (lane >= 16) ? 2 : 0;
    const int col = lane & 15;
    {   // layer 1: 234 -> 128, wave w handles output tile n0 = 16w
        const int n0 = wave * 16;
        v8f c; for (int q = 0; q < 8; ++q) c[q] = 0.f;
        for (int kt = 0; kt < 59; ++kt) {
            int k = kt * 4 + kb;
            v2f a; a.x = xv[k]; a.y = xv[k + 1];
            v2f b;
            b.x = (k     < 234) ? W1[(size_t)k * 128 + n0 + col]       : 0.f;
            b.y = (k + 1 < 234) ? W1[(size_t)(k + 1) * 128 + n0 + col] : 0.f;
            c = __builtin_amdgcn_wmma_f32_16x16x4_f32(false, a, false, b,
                                                      (short)0, c, false, false);
        }
        if (lane < 16) {
            float v = c[0] + b1[n0 + col];
            y1[n0 + col] = v > 0.f ? v : 0.f;
        }
    }
    __syncthreads();
    if (wave < 2) {   // layer 2: 128 -> 32
        const int n0 = wave * 16;
        v8f c; for (int q = 0; q < 8; ++q) c[q] = 0.f;
        for (int kt = 0; kt < 32; ++kt) {
            int k = kt * 4 + kb;
            v2f a; a.x = y1[k]; a.y = y1[k + 1];
            v2f b; b.x = W2[k * 32 + n0 + col]; b.y = W2[(k + 1) * 32 + n0 + col];
            c = __builtin_amdgcn_wmma_f32_16x16x4_f32(false, a, false, b,
                                                      (short)0, c, false, false);
        }
        if (lane < 16) {
            float v = c[0] + b2[n0 + col];
            y2[n0 + col] = v > 0.f ? v : 0.f;
        }
    }
    __syncthreads();
    if (tid == 0) {   // layer 3: 32 -> 1
        float s = b3[0];
        for (int i = 0; i < 32; ++i) s += y2[i] * W3[i];
        out[0] = s;
    }
}

// ---------------------------------------------------------------------------
extern "C" void kernel_launch(void* const* d_in, const int* in_sizes, int n_in,
                              void* d_out, int out_size, void* d_ws, size_t ws_size,
                              hipStream_t stream) {
    (void)in_sizes; (void)n_in; (void)out_size; (void)ws_size;
    const float* molfeats = (const float*)d_in[0];
    const float* h        = (const float*)d_in[2];
    const float* h2       = (const float*)d_in[3];
    const float* e        = (const float*)d_in[4];
    const int*   nbr      = (const int*)  d_in[5];
    const float* V1w = (const float*)d_in[6],  *V1b = (const float*)d_in[7];
    const float* E1w = (const float*)d_in[8],  *E1b = (const float*)d_in[9];
    const float* U1w = (const float*)d_in[10], *U1b = (const float*)d_in[11];
    const float* V2w = (const float*)d_in[12], *V2b = (const float*)d_in[13];
    const float* E2w = (const float*)d_in[14], *E2b = (const float*)d_in[15];
    const float* U2w = (const float*)d_in[16], *U2b = (const float*)d_in[17];
    const float* Rw  = (const float*)d_in[18], *Rb  = (const float*)d_in[19];
    const float* W1  = (const float*)d_in[20], *b1  = (const float*)d_in[21];
    const float* W2  = (const float*)d_in[22], *b2  = (const float*)d_in[23];
    const float* W3  = (const float*)d_in[24], *b3  = (const float*)d_in[25];

    float* ws  = (float*)d_ws;
    float* wse = ws + WSE_OFF;
    float* wsg = ws + WSG_OFF;
    float* wsh = ws + WSH_OFF;

    k_reduce_e     <<<K1_BLOCKS, K1_THREADS, 0, stream>>>((const v4f*)e, wse);
    k_reduce_gather<<<K2_BLOCKS, 256, 0, stream>>>(nbr, h, wsg);
    k_reduce_h     <<<K3_BLOCKS, 256, 0, stream>>>(h, h2, wsh);
    k_final<<<1, 256, 0, stream>>>(molfeats,
                                   V1w, V1b, E1w, E1b, U1w, U1b,
                                   V2w, V2b, E2w, E2b, U2w, U2b,
                                   Rw, Rb, W1, b1, W2, b2, W3, b3,
                                   wse, wsg, wsh, (float*)d_out);
}